// GATCN_89172111000293
// MI455X (gfx1250) — compile-verified
//
#include <hip/hip_runtime.h>
#include <hip/hip_bf16.h>

typedef float v2f __attribute__((ext_vector_type(2)));
typedef float v8f __attribute__((ext_vector_type(8)));
typedef unsigned int u32x4 __attribute__((ext_vector_type(4)));
typedef int i32x4 __attribute__((ext_vector_type(4)));
typedef int i32x8 __attribute__((ext_vector_type(8)));

#define G_DIM  256
#define H1     128
#define H2     128
#define NREL   8
#define NBASES 30

#if defined(__has_builtin)
#if __has_builtin(__builtin_amdgcn_tensor_load_to_lds) && \
    __has_builtin(__builtin_amdgcn_s_wait_tensorcnt)
#define USE_TDM 1
#endif
#endif
#ifndef USE_TDM
#define USE_TDM 0
#endif

// B matrices are stored K-pair-interleaved: element (k,col) of a [K,128]
// matrix lives at Bp[(k/2)*256 + col*2 + (k&1)].  A WMMA B-fragment
// {B[k][col], B[k+1][col]} (k even) is then one aligned 8-byte load.
__host__ __device__ __forceinline__ int bpack_idx(int k, int col) {
  return (k >> 1) * 256 + col * 2 + (k & 1);
}

// ---------------------------------------------------------------------------
// utility kernels
// ---------------------------------------------------------------------------
__global__ void zero_u32_kernel(unsigned* __restrict__ p, int n) {
  int i = blockIdx.x * blockDim.x + threadIdx.x;
  if (i < n) p[i] = 0u;
}

// Wr[r][k][o] = sum_b comp[r,b] * basis[b,k,o], stored pair-interleaved.
__global__ void wr_kernel(const float* __restrict__ comp,
                          const float* __restrict__ basis,
                          float* __restrict__ Wr, int total) {
  int i = blockIdx.x * blockDim.x + threadIdx.x;
  if (i >= total) return;
  int r  = i >> 15;        // / (256*128)
  int io = i & 32767;
  int k  = io >> 7;
  int o  = io & 127;
  float s = 0.f;
  #pragma unroll 5
  for (int b = 0; b < NBASES; ++b)
    s = fmaf(comp[r * NBASES + b], basis[b * (G_DIM * H1) + io], s);
  Wr[(r << 15) + bpack_idx(k, o)] = s;
}

// repack a [K,128] row-major matrix into the pair-interleaved layout
__global__ void pack_b_kernel(const float* __restrict__ B,
                              float* __restrict__ Bp, int total) {
  int i = blockIdx.x * blockDim.x + threadIdx.x;
  if (i >= total) return;
  int k = i >> 7;
  int o = i & 127;
  Bp[bpack_idx(k, o)] = B[i];
}

__global__ void count_kernel(const int* __restrict__ dst,
                             const int* __restrict__ etype,
                             unsigned* __restrict__ cnt, int E) {
  int e = blockIdx.x * blockDim.x + threadIdx.x;
  if (e < E) atomicAdd(&cnt[dst[e] * NREL + etype[e]], 1u);
}

__global__ void invcnt_kernel(const unsigned* __restrict__ cnt,
                              float* __restrict__ inv, int n) {
  int i = blockIdx.x * blockDim.x + threadIdx.x;
  if (i < n) inv[i] = 1.0f / fmaxf((float)cnt[i], 1.0f);
}

__global__ void init_out_kernel(float* __restrict__ out,
                                const float* __restrict__ bias, int n) {
  int i = blockIdx.x * blockDim.x + threadIdx.x;
  if (i < n) out[i] = bias[i & (H2 - 1)];
}

// ---------------------------------------------------------------------------
// TDM: async load of a contiguous [KC x 128] f32 panel into LDS at lds_off.
// Issued by wave 0 only; tracked with TENSORcnt.
// ---------------------------------------------------------------------------
#define BM 128
#define KC 32

#if USE_TDM
__device__ __forceinline__ void tdm_load_b_panel(const float* gptr,
                                                 unsigned lds_off) {
  unsigned long long ga = (unsigned long long)(uintptr_t)gptr;
  u32x4 g0;
  g0.x = 1u;                                         // count=1, user D#
  g0.y = lds_off;                                    // lds_addr (bytes)
  g0.z = (unsigned)(ga & 0xFFFFFFFFu);               // global_addr[31:0]
  g0.w = (unsigned)((ga >> 32) & 0x01FFFFFFu)        // global_addr[56:32]
       | (2u << 30);                                 // type = 2 (image)
  i32x8 g1;
  g1[0] = (2 << 16);            // workgroup_mask=0 (not in cluster), data_size=4B
  g1[1] = (128 << 16);          // tensor_dim0[15:0]=128 at bits 63:48
  g1[2] = (KC << 16);           // tensor_dim0[31:16]=0 | tensor_dim1[15:0]=KC
  g1[3] = (128 << 16);          // tensor_dim1[31:16]=0 | tile_dim0=128
  g1[4] = KC;                   // tile_dim1=KC, tile_dim2=0
  g1[5] = 128;                  // tensor_dim0_stride[31:0] = 128 elements
  g1[6] = 0;                    // stride0[47:32]=0 | tensor_dim1_stride[15:0]=0
  g1[7] = 0;
  i32x4 gz4 = {0, 0, 0, 0};             // groups 2/3 unused (2D tensor)
  i32x8 gz8 = {0, 0, 0, 0, 0, 0, 0, 0}; // extended group (clang-23 6-arg form)
  __builtin_amdgcn_tensor_load_to_lds(g0, g1, gz4, gz4, gz8, 0);
}
#endif

// ---------------------------------------------------------------------------
// fp32 WMMA GEMM: C[M,128] = A[M,K] @ B[K,128] (+bias), B pair-interleaved.
// block = 256 threads = 8 waves, each wave = 16 rows x 128 cols (8 WMMA tiles)
// B panel double-buffered in LDS, fetched by the Tensor Data Mover.
// ---------------------------------------------------------------------------
__global__ __launch_bounds__(256)
void gemm_wmma_f32(const float* __restrict__ A, int lda,
                   const float* __restrict__ B,     // [K,128] pair-interleaved
                   float* __restrict__ C, int ldc,
                   const float* __restrict__ bias,  // may be null
                   int M, int K) {
  __shared__ float sB[2][KC * 128];         // 2 x 16 KB
  const int tid  = threadIdx.x;
  const int wave = tid >> 5;
  const int lane = tid & 31;
  const int lo   = lane & 15;
  const int hi   = lane >> 4;               // K-half select for 16x16x4 frags
  const int m0   = blockIdx.x * BM + wave * 16;
  // clamp instead of predicate: tail garbage only pollutes rows we never store
  const int arow = min(m0 + lo, M - 1);
  const float* __restrict__ Arow = A + (size_t)arow * lda;

  v8f acc[8];
  #pragma unroll
  for (int t = 0; t < 8; ++t)
    #pragma unroll
    for (int j = 0; j < 8; ++j) acc[t][j] = 0.0f;

  const int nch = K / KC;                   // K is a multiple of KC here

#if USE_TDM
  if (wave == 0)
    tdm_load_b_panel(B, (unsigned)(uintptr_t)&sB[0][0]);
#endif

  for (int c = 0; c < nch; ++c) {
    const int k0 = c * KC;

    // preload this chunk's 8 A fragments as one clause (overlaps the barrier)
    v2f afrag[8];
    #pragma unroll
    for (int q = 0; q < 8; ++q) {
      float2 av = *(const float2*)(Arow + k0 + q * 4 + hi * 2);
      afrag[q].x = av.x; afrag[q].y = av.y;
    }
    if (k0 + KC < K)                        // global_prefetch_b8 for next A
      __builtin_prefetch(Arow + k0 + KC, 0, 1);

#if USE_TDM
    if (wave == 0) __builtin_amdgcn_s_wait_tensorcnt(0);
    __syncthreads();                        // buf(c&1) ready, buf((c+1)&1) free
    if (wave == 0 && c + 1 < nch)
      tdm_load_b_panel(B + (size_t)(k0 + KC) * 128,
                       (unsigned)(uintptr_t)&sB[(c + 1) & 1][0]);
    const float* __restrict__ sBc = &sB[c & 1][0];
#else
    __syncthreads();
    {
      const float4* gB  = (const float4*)(B + (size_t)k0 * 128);
      float4*       sB4 = (float4*)&sB[0][0];
      #pragma unroll
      for (int i = 0; i < 4; ++i) sB4[tid + i * 256] = gB[tid + i * 256];
    }
    __syncthreads();
    const float* __restrict__ sBc = &sB[0][0];
#endif

    #pragma unroll
    for (int kk = 0; kk < KC; kk += 4) {
      const v2f a = afrag[kk >> 2];
      // one aligned 8B LDS load per B fragment (pair-interleaved layout);
      // preload all 8 so the 8 WMMAs issue behind a single dscnt wait
      v2f bfrag[8];
      const float* sRow = sBc + ((kk + hi * 2) >> 1) * 256 + lo * 2;
      #pragma unroll
      for (int t = 0; t < 8; ++t)
        bfrag[t] = *(const v2f*)(sRow + t * 32);
      #pragma unroll
      for (int t = 0; t < 8; ++t)
        acc[t] = __builtin_amdgcn_wmma_f32_16x16x4_f32(
            /*neg_a=*/false, a, /*neg_b=*/false, bfrag[t],
            /*c_mod=*/(short)0, acc[t], /*reuse_a=*/false, /*reuse_b=*/false);
    }
  }

  // epilogue: VGPR v -> row m0 + v + 8*hi, col t*16 + lo
  float bcol[8];
  #pragma unroll
  for (int t = 0; t < 8; ++t) bcol[t] = bias ? bias[t * 16 + lo] : 0.0f;
  #pragma unroll
  for (int v = 0; v < 8; ++v) {
    const int r = m0 + v + hi * 8;
    if (r < M) {
      #pragma unroll
      for (int t = 0; t < 8; ++t)
        C[(size_t)r * ldc + t * 16 + lo] = acc[t][v] + bcol[t];
    }
  }
}

// ---------------------------------------------------------------------------
// RGCN scatter: one wave per edge, agg[dst] += xr[src] * inv_cnt[dst,rel]
// ---------------------------------------------------------------------------
__global__ void rgcn_scatter_kernel(const float* __restrict__ xr,
                                    const int* __restrict__ src,
                                    const int* __restrict__ dst,
                                    const int* __restrict__ etype,
                                    const float* __restrict__ invc,
                                    float* __restrict__ agg, int E, int rel) {
  int e = blockIdx.x * (blockDim.x >> 5) + (threadIdx.x >> 5);
  if (e >= E) return;
  if (etype[e] != rel) return;                  // wave-uniform predicate
  int lane = threadIdx.x & 31;
  int s = src[e], d = dst[e];
  float w = invc[d * NREL + rel];
  #pragma unroll
  for (int j = 0; j < 4; ++j) {
    int o = lane + j * 32;
    atomicAdd(&agg[(size_t)d * H1 + o], xr[(size_t)s * H1 + o] * w);
  }
}

// ---------------------------------------------------------------------------
// GATv2: scores + segment max (monotone-uint atomicMax)
// ---------------------------------------------------------------------------
__device__ __forceinline__ unsigned f2ord(float f) {
  unsigned u = __float_as_uint(f);
  return (u & 0x80000000u) ? ~u : (u | 0x80000000u);
}
__device__ __forceinline__ float ord2f(unsigned o) {
  unsigned u = (o & 0x80000000u) ? (o & 0x7FFFFFFFu) : ~o;
  return __uint_as_float(u);
}

__global__ void gat_score_kernel(const float* __restrict__ xl,
                                 const float* __restrict__ xrt,
                                 const float* __restrict__ att,
                                 const int* __restrict__ src,
                                 const int* __restrict__ dst,
                                 int E, int total,
                                 float* __restrict__ esc,
                                 unsigned* __restrict__ emax) {
  int idx = blockIdx.x * (blockDim.x >> 5) + (threadIdx.x >> 5);
  if (idx >= total) return;
  int lane = threadIdx.x & 31;
  int s, d;
  if (idx < E) { s = src[idx]; d = dst[idx]; } else { s = d = idx - E; }
  float partial = 0.f;
  #pragma unroll
  for (int j = 0; j < 4; ++j) {
    int o = lane + j * 32;
    float v = xl[(size_t)s * H2 + o] + xrt[(size_t)d * H2 + o];
    v = (v > 0.f) ? v : 0.2f * v;               // leaky_relu(0.2)
    partial = fmaf(v, att[o], partial);
  }
  #pragma unroll
  for (int off = 16; off; off >>= 1) partial += __shfl_xor(partial, off, 32);
  if (lane == 0) {
    esc[idx] = partial;
    atomicMax(&emax[d], f2ord(partial));
  }
}

__global__ void gat_exp_kernel(float* __restrict__ esc,
                               const unsigned* __restrict__ emax,
                               float* __restrict__ denom,
                               const int* __restrict__ dst,
                               int E, int total) {
  int idx = blockIdx.x * blockDim.x + threadIdx.x;
  if (idx >= total) return;
  int d = (idx < E) ? dst[idx] : (idx - E);
  float ex = __expf(esc[idx] - ord2f(emax[d]));
  esc[idx] = ex;
  atomicAdd(&denom[d], ex);
}

__global__ void gat_scatter_kernel(const float* __restrict__ esc,
                                   const float* __restrict__ denom,
                                   const float* __restrict__ xl,
                                   const int* __restrict__ src,
                                   const int* __restrict__ dst,
                                   int E, int total,
                                   float* __restrict__ out) {
  int idx = blockIdx.x * (blockDim.x >> 5) + (threadIdx.x >> 5);
  if (idx >= total) return;
  int lane = threadIdx.x & 31;
  int s, d;
  if (idx < E) { s = src[idx]; d = dst[idx]; } else { s = d = idx - E; }
  float alpha = esc[idx] / denom[d];
  #pragma unroll
  for (int j = 0; j < 4; ++j) {
    int o = lane + j * 32;
    atomicAdd(&out[(size_t)d * H2 + o], alpha * xl[(size_t)s * H2 + o]);
  }
}

// ---------------------------------------------------------------------------
// host side
// ---------------------------------------------------------------------------
extern "C" void kernel_launch(void* const* d_in, const int* in_sizes, int n_in,
                              void* d_out, int out_size, void* d_ws, size_t ws_size,
                              hipStream_t stream) {
  const float* x     = (const float*)d_in[0];
  const int*   eidx  = (const int*)d_in[1];
  // d_in[2] = edge_norm (unused, per reference)
  const int*   etype = (const int*)d_in[3];
  const float* basis = (const float*)d_in[4];
  const float* comp  = (const float*)d_in[5];
  const float* root  = (const float*)d_in[6];
  const float* bias1 = (const float*)d_in[7];
  const float* w_l   = (const float*)d_in[8];
  const float* w_r   = (const float*)d_in[9];
  const float* att   = (const float*)d_in[10];
  const float* bias2 = (const float*)d_in[11];
  float* out = (float*)d_out;

  const int N = in_sizes[0] / G_DIM;
  const int E = in_sizes[3];
  const int* src = eidx;
  const int* dst = eidx + E;
  const int total = E + N;

  // workspace carve-up (256B aligned regions)
  char* ws = (char*)d_ws;
  size_t off = 0;
  auto carve = [&](size_t bytes) -> char* {
    char* p = ws + off;
    off += (bytes + 255) & ~(size_t)255;
    return p;
  };
  float*    Wr    = (float*)carve(sizeof(float) * NREL * G_DIM * H1);   // packed
  float*    rootP = (float*)carve(sizeof(float) * G_DIM * H1);          // packed
  float*    wlP   = (float*)carve(sizeof(float) * H1 * H2);             // packed
  float*    wrP   = (float*)carve(sizeof(float) * H1 * H2);             // packed
  float*    xr    = (float*)carve(sizeof(float) * (size_t)N * H1);      // per-relation
  float*    agg   = (float*)carve(sizeof(float) * (size_t)N * H1);      // becomes x1
  unsigned* cnt   = (unsigned*)carve(sizeof(unsigned) * (size_t)N * NREL);
  float*    invc  = (float*)carve(sizeof(float) * (size_t)N * NREL);
  float*    xl    = (float*)carve(sizeof(float) * (size_t)N * H2);
  float*    xrt   = (float*)carve(sizeof(float) * (size_t)N * H2);
  float*    esc   = (float*)carve(sizeof(float) * (size_t)total);
  unsigned* emax  = (unsigned*)carve(sizeof(unsigned) * (size_t)N);
  float*    denom = (float*)carve(sizeof(float) * (size_t)N);
  (void)ws_size; (void)n_in; (void)out_size;

  const dim3 gemmGrid((N + BM - 1) / BM);

  // ---- stage 1: RGCN ----
  const int wrTotal = NREL * G_DIM * H1;
  wr_kernel<<<(wrTotal + 255) / 256, 256, 0, stream>>>(comp, basis, Wr, wrTotal);
  pack_b_kernel<<<(G_DIM * H1 + 255) / 256, 256, 0, stream>>>(root, rootP, G_DIM * H1);
  pack_b_kernel<<<(H1 * H2 + 255) / 256, 256, 0, stream>>>(w_l, wlP, H1 * H2);
  pack_b_kernel<<<(H1 * H2 + 255) / 256, 256, 0, stream>>>(w_r, wrP, H1 * H2);

  // agg = x @ root + bias1
  gemm_wmma_f32<<<gemmGrid, 256, 0, stream>>>(x, G_DIM, rootP, agg, H1, bias1, N, G_DIM);

  zero_u32_kernel<<<(N * NREL + 255) / 256, 256, 0, stream>>>(cnt, N * NREL);
  count_kernel<<<(E + 255) / 256, 256, 0, stream>>>(dst, etype, cnt, E);
  invcnt_kernel<<<(N * NREL + 255) / 256, 256, 0, stream>>>(cnt, invc, N * NREL);

  for (int r = 0; r < NREL; ++r) {
    gemm_wmma_f32<<<gemmGrid, 256, 0, stream>>>(
        x, G_DIM, Wr + (size_t)r * G_DIM * H1, xr, H1, nullptr, N, G_DIM);
    rgcn_scatter_kernel<<<(E + 7) / 8, 256, 0, stream>>>(
        xr, src, dst, etype, invc, agg, E, r);
  }

  // ---- stage 2: GATv2 ----
  gemm_wmma_f32<<<gemmGrid, 256, 0, stream>>>(agg, H1, wlP, xl,  H2, nullptr, N, H1);
  gemm_wmma_f32<<<gemmGrid, 256, 0, stream>>>(agg, H1, wrP, xrt, H2, nullptr, N, H1);

  zero_u32_kernel<<<(N + 255) / 256, 256, 0, stream>>>(emax, N);
  zero_u32_kernel<<<(N + 255) / 256, 256, 0, stream>>>((unsigned*)denom, N);

  gat_score_kernel<<<(total + 7) / 8, 256, 0, stream>>>(
      xl, xrt, att, src, dst, E, total, esc, emax);
  gat_exp_kernel<<<(total + 255) / 256, 256, 0, stream>>>(
      esc, emax, denom, dst, E, total);
  init_out_kernel<<<(N * H2 + 255) / 256, 256, 0, stream>>>(out, bias2, N * H2);
  gat_scatter_kernel<<<(total + 7) / 8, 256, 0, stream>>>(
      esc, denom, xl, src, dst, E, total, out);
}